// S4Model_24575802867891
// MI455X (gfx1250) — compile-verified
//
#include <hip/hip_runtime.h>
#include <math.h>

typedef __attribute__((ext_vector_type(2))) float v2f;
typedef __attribute__((ext_vector_type(8))) float v8f;

#define SRATE 30225.0f
#define B_    16
#define LIN   60450
#define L1    60200
#define L2    7524
#define L3    3760
#define H_    128
#define N_    32
#define PI_F  3.14159265358979323846f

#define NT_SINC 3763   // ceil(L1/16)
#define NT_C1   471    // ceil(L2/16)
#define NT_C2   235    // L3/16 exact

// ---------------------------------------------------------------- sinc filters (padded 16x252)
__global__ void k_sinc_filt(const float* __restrict__ low_hz,
                            const float* __restrict__ band_hz,
                            float* __restrict__ filt) {
  int idx = blockIdx.x * 256 + threadIdx.x;
  if (idx >= 16 * 252) return;
  int c = idx / 252, k = idx % 252;
  float v;
  if (k == 251) {
    v = 0.f;                                // zero pad tap so K%4==0
  } else if (k == 125) {
    v = 1.0f;                               // 2*band / (2*band)
  } else {
    float low  = 50.f + fabsf(low_hz[c]);
    float high = fminf(fmaxf(low + 50.f + fabsf(band_hz[c]), 50.f), SRATE * 0.5f);
    float band = high - low;
    int i = (k < 125) ? k : (250 - k);
    float nlin = (float)i * (124.5f / 124.f);           // linspace(0,124.5,125)
    float win  = 0.54f - 0.46f * cosf(2.f * PI_F * nlin / 251.f);
    float nn   = 2.f * PI_F * (float)(i - 125) / SRATE; // arange(-125,0)
    float bpl  = (sinf(high * nn) - sinf(low * nn)) / (0.5f * nn) * win;
    v = bpl / (2.f * band);
  }
  filt[idx] = v;
}

// ---------------------- one-shot weight permutes: im2col K reordered tap-major (r' = k*C + c)
__global__ void k_permw(const float* __restrict__ w1, const float* __restrict__ w2,
                        float* __restrict__ w1p, float* __restrict__ w2p) {
  int i = blockIdx.x * 256 + threadIdx.x;
  if (i < 64 * 240) {
    int o = i / 240, r = i % 240;          // r = k*16 + c
    int k = r >> 4, c = r & 15;
    w1p[i] = w1[o * 240 + c * 15 + k];
  }
  if (i < 128 * 384) {
    int o = i / 384, r = i % 384;          // r = k*64 + c
    int k = r >> 6, c = r & 63;
    w2p[i] = w2[o * 384 + c * 6 + k];
  }
}

// ------------------------------------------- sinc conv as WMMA: (16 x 252) @ (252 x L1)
// one wave per 16x16 tile; M=16, K=252 (63 steps). Fast path: pure immediate-offset loads.
__global__ void k_sincconv(const float* __restrict__ x,
                           const float* __restrict__ filt,
                           float* __restrict__ h1) {
  int gid  = blockIdx.x * 8 + (threadIdx.x >> 5);
  int b    = gid / NT_SINC;
  int nt   = gid - b * NT_SINC;
  int lane = threadIdx.x & 31;
  int half = lane >> 4;
  int lm   = lane & 15;
  const float* Ab = filt + lm * 252 + 2 * half;   // A row = channel lm
  int n = nt * 16 + lm;
  v8f acc = {};
  if (b == B_ - 1 && nt == NT_SINC - 1) {
    // slow tail: clamp loads to stay inside the x buffer (tail cols unstored; r=251 tap is 0)
    const float* Xb = x + b * LIN;
#pragma unroll
    for (int k0 = 0; k0 < 252; k0 += 4) {
      v2f a = *(const v2f*)(Ab + k0);
      int i0 = n + k0 + 2 * half;
      v2f bb;
      bb.x = Xb[min(i0, LIN - 1)];
      bb.y = Xb[min(i0 + 1, LIN - 1)];
      acc = __builtin_amdgcn_wmma_f32_16x16x4_f32(false, a, false, bb, (short)0, acc,
                                                  false, false);
    }
  } else {
    // fast path: base pointer + literal offsets k0, k0+1
    const float* Xp = x + b * LIN + n + 2 * half;
#pragma unroll
    for (int k0 = 0; k0 < 252; k0 += 4) {
      v2f a = *(const v2f*)(Ab + k0);
      v2f bb;
      bb.x = Xp[k0];
      bb.y = Xp[k0 + 1];
      acc = __builtin_amdgcn_wmma_f32_16x16x4_f32(false, a, false, bb, (short)0, acc,
                                                  false, false);
    }
  }
  if (n < L1) {
#pragma unroll
    for (int j = 0; j < 8; ++j) {
      int c = j + 8 * half;
      h1[(b * 16 + c) * L1 + n] = acc[j];
    }
  }
}

// ------------------- conv1 as WMMA: (64 x 240) @ im2col(240 x L2), s=8, tap-major K, BN+ReLU
__global__ void k_conv1(const float* __restrict__ h1, const float* __restrict__ w1p,
                        const float* __restrict__ g1, const float* __restrict__ b1,
                        const float* __restrict__ m1, const float* __restrict__ v1,
                        float* __restrict__ h2) {
  int gid  = blockIdx.x * 8 + (threadIdx.x >> 5);
  int b    = gid / (4 * NT_C1);
  int rem  = gid - b * (4 * NT_C1);
  int mt   = rem / NT_C1;
  int nt   = rem - mt * NT_C1;
  int lane = threadIdx.x & 31;
  int half = lane >> 4;
  int lm   = lane & 15;
  const float* Ab = w1p + (mt * 16 + lm) * 240 + 2 * half;
  int n  = nt * 16 + lm;
  int nc = min(n, L2 - 1);          // clamp (hoisted); tail cols unstored
  // K row r' = k*16 + c; within a 4-group, tap k = k0>>4 is a literal, channel = (k0&15)+2*half
  const float* P = h1 + b * 16 * L1 + 2 * half * L1 + 8 * nc;
  v8f acc = {};
#pragma unroll
  for (int k0 = 0; k0 < 240; k0 += 4) {
    v2f a = *(const v2f*)(Ab + k0);
    v2f bb;
    bb.x = P[(k0 & 15) * L1 + (k0 >> 4)];
    bb.y = P[((k0 & 15) + 1) * L1 + (k0 >> 4)];
    acc = __builtin_amdgcn_wmma_f32_16x16x4_f32(false, a, false, bb, (short)0, acc,
                                                false, false);
  }
  if (n < L2) {
#pragma unroll
    for (int j = 0; j < 8; ++j) {
      int o  = mt * 16 + j + 8 * half;
      float sc = g1[o] * rsqrtf(v1[o] + 1e-5f);
      float sh = b1[o] - m1[o] * sc;
      h2[(b * 64 + o) * L2 + n] = fmaxf(fmaf(acc[j], sc, sh), 0.f);
    }
  }
}

// ------------------- conv2 as WMMA: (128 x 384) @ im2col(384 x L3), s=2, tap-major K, BN+ReLU
__global__ void k_conv2(const float* __restrict__ h2, const float* __restrict__ w2p,
                        const float* __restrict__ g2, const float* __restrict__ b2,
                        const float* __restrict__ m2, const float* __restrict__ v2,
                        float* __restrict__ h) {
  int gid  = blockIdx.x * 8 + (threadIdx.x >> 5);
  int b    = gid / (8 * NT_C2);
  int rem  = gid - b * (8 * NT_C2);
  int mt   = rem / NT_C2;
  int nt   = rem - mt * NT_C2;
  int lane = threadIdx.x & 31;
  int half = lane >> 4;
  int lm   = lane & 15;
  const float* Ab = w2p + (mt * 16 + lm) * 384 + 2 * half;
  int n = nt * 16 + lm;            // exact tiling, always in-bounds
  // K row r' = k*64 + c; tap k = k0>>6 literal, channel = (k0&63)+2*half
  const float* P = h2 + b * 64 * L2 + 2 * half * L2 + 2 * n;
  v8f acc = {};
#pragma unroll
  for (int k0 = 0; k0 < 384; k0 += 4) {
    v2f a = *(const v2f*)(Ab + k0);
    v2f bb;
    bb.x = P[(k0 & 63) * L2 + (k0 >> 6)];
    bb.y = P[((k0 & 63) + 1) * L2 + (k0 >> 6)];
    acc = __builtin_amdgcn_wmma_f32_16x16x4_f32(false, a, false, bb, (short)0, acc,
                                                false, false);
  }
#pragma unroll
  for (int j = 0; j < 8; ++j) {
    int o  = mt * 16 + j + 8 * half;
    float sc = g2[o] * rsqrtf(v2[o] + 1e-5f);
    float sh = b2[o] - m2[o] * sc;
    h[(b * 128 + o) * L3 + n] = fmaxf(fmaf(acc[j], sc, sh), 0.f);
  }
}

// ---------------------------------------------------------------- S4D discretized params
__global__ void k_s4param(const float* __restrict__ log_dt, const float* __restrict__ C_re,
                          const float* __restrict__ C_im, const float* __restrict__ logA_re,
                          const float* __restrict__ A_im,
                          float* __restrict__ dare, float* __restrict__ daim,
                          float* __restrict__ cpre, float* __restrict__ cpim) {
  int i = blockIdx.x * 256 + threadIdx.x;   // H_*N_ = 4096 threads
  int hh = i >> 5;
  float dt   = expf(log_dt[hh]);
  float Are  = -expf(logA_re[i]);
  float Aim  = A_im[i];
  float dre  = Are * dt, dimv = Aim * dt;
  float ed   = expf(dre);
  float dAre = ed * cosf(dimv), dAim = ed * sinf(dimv);
  float nre  = dAre - 1.f, nim = dAim;
  float den  = Are * Are + Aim * Aim;
  float wre  = (nre * Are + nim * Aim) / den;   // (e^{dtA}-1)/A
  float wim  = (nim * Are - nre * Aim) / den;
  float Cr = C_re[i], Ci = C_im[i];
  dare[i] = dAre; daim[i] = dAim;
  cpre[i] = 2.f * (Cr * wre - Ci * wim);        // fold factor 2 into C'
  cpim[i] = 2.f * (Cr * wim + Ci * wre);
}

// ------------------------------------------------- S4D scan (lane = state n) + D skip + GELU
__global__ void k_scan(const float* __restrict__ u_all, const float* __restrict__ dare,
                       const float* __restrict__ daim, const float* __restrict__ cpre,
                       const float* __restrict__ cpim, const float* __restrict__ Dp,
                       float* __restrict__ yg) {
  int w    = blockIdx.x * 8 + (threadIdx.x >> 5);  // one wave per (b,h), 2048 waves
  int lane = threadIdx.x & 31;
  int hh   = w & (H_ - 1);
  const float* u = u_all + (size_t)w * L3;
  float* y       = yg   + (size_t)w * L3;
  int pi = hh * 32 + lane;
  float ar = dare[pi], ai = daim[pi], cr = cpre[pi], ci = cpim[pi];
  float Dh = Dp[hh];
  float sre = 0.f, sim = 0.f;
  for (int t0 = 0; t0 < L3; t0 += 32) {
    __builtin_prefetch(u + t0 + 64, 0, 0);
    int t = t0 + lane;
    float uv = (t < L3) ? u[t] : 0.f;
    int cnt = min(32, L3 - t0);
    float yv = 0.f;
    for (int j = 0; j < cnt; ++j) {
      float uj  = __shfl(uv, j, 32);
      float nre = fmaf(ar, sre, fmaf(-ai, sim, uj));
      float nim = fmaf(ar, sim, ai * sre);
      sre = nre; sim = nim;
      float p = cr * sre - ci * sim;
#pragma unroll
      for (int off = 16; off > 0; off >>= 1) p += __shfl_xor(p, off, 32);
      if (lane == j) yv = fmaf(Dh, uj, p);
    }
    if (t < L3) y[t] = 0.5f * yv * (1.f + erff(yv * 0.70710678118f));  // exact GELU
  }
}

// ---------------------------------------------------- z = Wo @ gelu(y) + bo via fp32 WMMA
// One wave per 16x16 output tile. M=256, N=3760=235*16, K=128=32*4 (all exact, EXEC all-ones).
__global__ void k_gemm(const float* __restrict__ Wo, const float* __restrict__ bo,
                       const float* __restrict__ G, float* __restrict__ Z) {
  int gid  = blockIdx.x * 8 + (threadIdx.x >> 5);
  int b    = gid / (16 * 235);
  int rem  = gid - b * (16 * 235);
  int mt   = rem / 235;
  int nt   = rem - mt * 235;
  int lane = threadIdx.x & 31;
  int half = lane >> 4;
  int lm   = lane & 15;
  const float* Ab = Wo + (mt * 16 + lm) * 128 + 2 * half;
  const float* Gb = G + b * (128 * L3) + nt * 16 + lm + 2 * half * L3;
  v8f acc = {};
#pragma unroll
  for (int k0 = 0; k0 < 128; k0 += 4) {
    v2f a = *(const v2f*)(Ab + k0);
    v2f bb;
    bb.x = Gb[k0 * L3];
    bb.y = Gb[k0 * L3 + L3];
    acc = __builtin_amdgcn_wmma_f32_16x16x4_f32(false, a, false, bb, (short)0, acc,
                                                false, false);
  }
  float* Zb = Z + b * (256 * L3) + nt * 16 + lm;
#pragma unroll
  for (int j = 0; j < 8; ++j) {
    int m = mt * 16 + j + 8 * half;   // C/D: VGPR j -> M=j (lo half) / j+8 (hi half)
    Zb[m * L3] = acc[j] + bo[m];
  }
}

// -------------------------------------- GLU + residual + LayerNorm(channels), in-place on h
__global__ void k_glu_ln(const float* __restrict__ Z, float* __restrict__ Hb,
                         const float* __restrict__ gam, const float* __restrict__ bet) {
  __shared__ float vals[128 * 32];
  __shared__ float psum[8 * 32];
  __shared__ float psq[8 * 32];
  __shared__ float smu[32], srs[32];
  int b  = blockIdx.y;
  int t0 = blockIdx.x * 32;
  int tx = threadIdx.x & 31;
  int ty = threadIdx.x >> 5;
  int t  = t0 + tx;
  bool valid = t < L3;
  const float* Zb = Z + b * (256 * L3) + t;
  float* Hp       = Hb + b * (128 * L3) + t;
  float sum = 0.f, sq = 0.f;
#pragma unroll
  for (int k = 0; k < 16; ++k) {
    int c = ty * 16 + k;
    float v = 0.f;
    if (valid) {
      float zt = Zb[c * L3];
      float zg = Zb[(c + 128) * L3];
      float hv = Hp[c * L3];
      v = fmaf(zt, 1.f / (1.f + expf(-zg)), hv);
    }
    vals[c * 32 + tx] = v;
    sum += v; sq += v * v;
  }
  psum[ty * 32 + tx] = sum;
  psq[ty * 32 + tx]  = sq;
  __syncthreads();
  if (ty == 0) {
    float s = 0.f, q = 0.f;
#pragma unroll
    for (int y = 0; y < 8; ++y) { s += psum[y * 32 + tx]; q += psq[y * 32 + tx]; }
    float mu  = s * (1.f / 128.f);
    float var = q * (1.f / 128.f) - mu * mu;
    smu[tx] = mu;
    srs[tx] = rsqrtf(var + 1e-5f);
  }
  __syncthreads();
  float mu = smu[tx], rs = srs[tx];
#pragma unroll
  for (int k = 0; k < 16; ++k) {
    int c = ty * 16 + k;
    if (valid) Hp[c * L3] = (vals[c * 32 + tx] - mu) * rs * gam[c] + bet[c];
  }
}

// ---------------------------------------------------------------- mean pool + head
__global__ void k_head(const float* __restrict__ Hb, const float* __restrict__ Wd,
                       const float* __restrict__ bd, float* __restrict__ out) {
  __shared__ float pooled[128];
  int b = blockIdx.x;
  int c = threadIdx.x;
  const float* p = Hb + (size_t)(b * 128 + c) * L3;
  float s = 0.f;
  for (int t = 0; t < L3; ++t) s += p[t];
  pooled[c] = s * (1.f / L3);
  __syncthreads();
  if (c < 10) {
    float acc = bd[c];
    for (int k = 0; k < 128; ++k) acc = fmaf(pooled[k], Wd[c * 128 + k], acc);
    out[b * 10 + c] = acc;
  }
}

// ---------------------------------------------------------------- launch
extern "C" void kernel_launch(void* const* d_in, const int* in_sizes, int n_in,
                              void* d_out, int out_size, void* d_ws, size_t ws_size,
                              hipStream_t stream) {
  (void)in_sizes; (void)n_in; (void)out_size; (void)ws_size;
  const float* x       = (const float*)d_in[0];
  const float* low_hz  = (const float*)d_in[1];
  const float* band_hz = (const float*)d_in[2];
  const float* w1      = (const float*)d_in[3];
  const float* g1      = (const float*)d_in[4];
  const float* b1      = (const float*)d_in[5];
  const float* m1      = (const float*)d_in[6];
  const float* v1      = (const float*)d_in[7];
  const float* w2      = (const float*)d_in[8];
  const float* g2      = (const float*)d_in[9];
  const float* b2      = (const float*)d_in[10];
  const float* m2      = (const float*)d_in[11];
  const float* v2      = (const float*)d_in[12];
  const float* log_dt  = (const float*)d_in[13];
  const float* C_re    = (const float*)d_in[14];
  const float* C_im    = (const float*)d_in[15];
  const float* logA_re = (const float*)d_in[16];
  const float* A_im    = (const float*)d_in[17];
  const float* Dp      = (const float*)d_in[18];
  const float* Wo      = (const float*)d_in[19];
  const float* bo      = (const float*)d_in[20];
  const float* ln_g    = (const float*)d_in[21];
  const float* ln_b    = (const float*)d_in[22];
  const float* Wd      = (const float*)d_in[23];
  const float* bd      = (const float*)d_in[24];
  float* out = (float*)d_out;
  float* ws  = (float*)d_ws;

  // workspace layout (floats); z buffer reuses the (dead) sinc-conv h1 region
  float* filt = ws;                                   // 16*252 = 4032 (<4096)
  float* dare = ws + 4096;
  float* daim = ws + 8192;
  float* cpre = ws + 12288;
  float* cpim = ws + 16384;
  float* w1p  = ws + 20480;                           // 15360 (pad to 16384)
  float* w2p  = ws + 36864;                           // 49152
  float* hbuf = ws + 86016;                           // B*H*L3 = 7,700,480
  float* ygel = hbuf + (size_t)B_ * H_ * L3;          // B*H*L3
  float* zbuf = ygel + (size_t)B_ * H_ * L3;          // max(B*256*L3, B*16*L1) = 15,411,200
  float* h1   = zbuf;
  float* h2   = zbuf + (size_t)B_ * 16 * L1;          // B*64*L2 = 7,704,576

  k_sinc_filt<<<16, 256, 0, stream>>>(low_hz, band_hz, filt);
  k_permw<<<192, 256, 0, stream>>>(w1, w2, w1p, w2p);
  k_sincconv<<<(B_ * NT_SINC) / 8, 256, 0, stream>>>(x, filt, h1);
  k_conv1<<<(B_ * 4 * NT_C1) / 8, 256, 0, stream>>>(h1, w1p, g1, b1, m1, v1, h2);
  k_conv2<<<(B_ * 8 * NT_C2) / 8, 256, 0, stream>>>(h2, w2p, g2, b2, m2, v2, hbuf);

  for (int l = 0; l < 4; ++l) {
    k_s4param<<<16, 256, 0, stream>>>(log_dt + l * H_, C_re + l * H_ * N_,
                                      C_im + l * H_ * N_, logA_re + l * H_ * N_,
                                      A_im + l * H_ * N_, dare, daim, cpre, cpim);
    k_scan<<<256, 256, 0, stream>>>(hbuf, dare, daim, cpre, cpim, Dp + l * H_, ygel);
    k_gemm<<<(16 * 16 * 235) / 8, 256, 0, stream>>>(Wo + l * 256 * H_, bo + l * 256,
                                                    ygel, zbuf);
    k_glu_ln<<<dim3((L3 + 31) / 32, B_), 256, 0, stream>>>(zbuf, hbuf,
                                                           ln_g + l * H_, ln_b + l * H_);
  }
  k_head<<<B_, 128, 0, stream>>>(hbuf, Wd, bd, out);
}